// EAGLE3DraftANE_4355096838297
// MI455X (gfx1250) — compile-verified
//
#include <hip/hip_runtime.h>
#include <hip/hip_fp16.h>

// EAGLE3 draft step for MI455X (gfx1250, wave32).
// Bandwidth-bound GEMV pipeline: ~1.45 GB fp32 weights -> ~62us floor @23.3TB/s.
// Matrix math via V_WMMA_F32_16X16X4_F32 (16-row tiles per wave), 128-bit
// global loads of W, activation vector staged in LDS (ds_load_b128).
// Dead q/k projections from the reference are skipped (not in output path).

#define H_     4096
#define NKV_   8
#define HD_    128
#define FFN_   14336
#define VOCAB_ 32000
#define EPS_   1e-6f

typedef __attribute__((ext_vector_type(2))) float v2f;
typedef __attribute__((ext_vector_type(8))) float v8f;

// One 8-wide K-chunk: two f32 WMMAs over permuted K-sets {k,k+1,k+4,k+5} and
// {k+2,k+3,k+6,k+7}. lane(m,h): w4 = W[row0+m][k+4h..k+4h+3], x4 = x[k+4h..].
__device__ __forceinline__ v8f wmma_step(const float4 w4, const float4 x4, v8f c) {
  v2f a0, b0, a1, b1;
  a0[0] = w4.x; a0[1] = w4.y; b0[0] = x4.x; b0[1] = x4.y;
  c = __builtin_amdgcn_wmma_f32_16x16x4_f32(false, a0, false, b0, (short)0, c, false, false);
  a1[0] = w4.z; a1[1] = w4.w; b1[0] = x4.z; b1[1] = x4.w;
  c = __builtin_amdgcn_wmma_f32_16x16x4_f32(false, a1, false, b1, (short)0, c, false, false);
  return c;
}

__device__ __forceinline__ v8f gemv_tile(const float* __restrict__ W,
                                         const float* __restrict__ sx,
                                         int row0, int K) {
  const int lane = threadIdx.x & 31;
  const int m = lane & 15;
  const int h = (lane >> 4) & 1;
  const float* wrow = W + (size_t)(row0 + m) * (size_t)K;
  v8f c = {};
  for (int k = 0; k < K; k += 8) {
    float4 w4 = *(const float4*)(wrow + k + 4 * h);
    float4 x4 = *(const float4*)(sx + k + 4 * h);
    c = wmma_step(w4, x4, c);
  }
  return c;
}

// D columns are all identical (B is broadcast vector): lane 0 holds rows
// row0..row0+7 in c[0..7], lane 16 holds rows row0+8..row0+15.
__device__ __forceinline__ void store_tile(float* __restrict__ out,
                                           const float* __restrict__ resid,
                                           int row0, v8f c) {
  const int lane = threadIdx.x & 31;
  if ((lane & 15) == 0) {
    int base = row0 + (lane >> 4) * 8;
#pragma unroll
    for (int g = 0; g < 8; ++g) {
      float acc = c[g];
      if (resid) acc += resid[base + g];
      out[base + g] = acc;
    }
  }
}

// Generic GEMV: out[r] = dot(W[r,:], xhat) (+resid). xhat built in LDS with
// optional RMS scale (inv), elementwise gain, and GQA repeat-gather.
__global__ void k_gemv(const float* __restrict__ W, const float* __restrict__ xsrc,
                       const float* __restrict__ gain, const float* __restrict__ inv,
                       const float* __restrict__ resid, int gqa,
                       float* __restrict__ out, int R, int K) {
  extern __shared__ float sx[];
  const float s = inv ? inv[0] : 1.0f;
  for (int k = threadIdx.x; k < K; k += blockDim.x) {
    float val = gqa ? xsrc[((k >> 9) << 7) + (k & 127)] : xsrc[k];
    if (gain) val *= gain[k];
    sx[k] = val * s;
  }
  __syncthreads();
  const int wave = threadIdx.x >> 5;
  const int row0 = (blockIdx.x * 8 + wave) * 16;
  if (row0 >= R) return;  // wave-uniform
  v8f c = gemv_tile(W, sx, row0, K);
  store_tile(out, resid, row0, c);
}

// x[r] = dot(W1[r],x1) + dot(W2[r],x2) — single accumulator, chained WMMAs.
__global__ void k_dual_gemv(const float* __restrict__ W1, const float* __restrict__ W2,
                            const float* __restrict__ x1, const float* __restrict__ x2,
                            float* __restrict__ out, int R, int K) {
  extern __shared__ float sx[];
  float* sa = sx;
  float* sb = sx + K;
  for (int k = threadIdx.x; k < K; k += blockDim.x) { sa[k] = x1[k]; sb[k] = x2[k]; }
  __syncthreads();
  const int wave = threadIdx.x >> 5;
  const int row0 = (blockIdx.x * 8 + wave) * 16;
  if (row0 >= R) return;
  const int lane = threadIdx.x & 31;
  const int m = lane & 15;
  const int h = (lane >> 4) & 1;
  const float* w1 = W1 + (size_t)(row0 + m) * (size_t)K;
  const float* w2 = W2 + (size_t)(row0 + m) * (size_t)K;
  v8f c = {};
  for (int k = 0; k < K; k += 8) {
    c = wmma_step(*(const float4*)(w1 + k + 4 * h), *(const float4*)(sa + k + 4 * h), c);
    c = wmma_step(*(const float4*)(w2 + k + 4 * h), *(const float4*)(sb + k + 4 * h), c);
  }
  store_tile(out, nullptr, row0, c);
}

// Fused SwiGLU: t[r] = silu(dot(Wg[r],xhat)) * dot(Wu[r],xhat)
__global__ void k_gateup(const float* __restrict__ Wg, const float* __restrict__ Wu,
                         const float* __restrict__ x, const float* __restrict__ gain,
                         const float* __restrict__ inv, float* __restrict__ out,
                         int R, int K) {
  extern __shared__ float sx[];
  const float s = inv[0];
  for (int k = threadIdx.x; k < K; k += blockDim.x) sx[k] = x[k] * gain[k] * s;
  __syncthreads();
  const int wave = threadIdx.x >> 5;
  const int row0 = (blockIdx.x * 8 + wave) * 16;
  if (row0 >= R) return;
  const int lane = threadIdx.x & 31;
  const int m = lane & 15;
  const int h = (lane >> 4) & 1;
  const float* wg = Wg + (size_t)(row0 + m) * (size_t)K;
  const float* wu = Wu + (size_t)(row0 + m) * (size_t)K;
  v8f cg = {}, cu = {};
  for (int k = 0; k < K; k += 8) {
    float4 xv = *(const float4*)(sx + k + 4 * h);
    cg = wmma_step(*(const float4*)(wg + k + 4 * h), xv, cg);
    cu = wmma_step(*(const float4*)(wu + k + 4 * h), xv, cu);
  }
  if ((lane & 15) == 0) {
    int base = row0 + (lane >> 4) * 8;
#pragma unroll
    for (int g = 0; g < 8; ++g) {
      float gv = cg[g], uv = cu[g];
      out[base + g] = gv / (1.0f + __expf(-gv)) * uv;
    }
  }
}

// inv[0] = rsqrt(mean(x^2) + eps)
__global__ void k_rms_reduce(const float* __restrict__ x, int n, float* __restrict__ inv) {
  __shared__ float sred[32];
  float acc = 0.f;
  for (int i = threadIdx.x; i < n; i += blockDim.x) { float v = x[i]; acc += v * v; }
  for (int o = 16; o > 0; o >>= 1) acc += __shfl_down(acc, o, 32);
  const int wid = threadIdx.x >> 5;
  if ((threadIdx.x & 31) == 0) sred[wid] = acc;
  __syncthreads();
  if (threadIdx.x < 32) {
    const int nw = (blockDim.x + 31) >> 5;
    float a = (threadIdx.x < nw) ? sred[threadIdx.x] : 0.f;
    for (int o = 16; o > 0; o >>= 1) a += __shfl_down(a, o, 32);
    if (threadIdx.x == 0) inv[0] = rsqrtf(a / (float)n + EPS_);
  }
}

// Per-head RMSNormNoScale of v, in place. grid=NKV, block=HD.
__global__ void k_headnorm(float* __restrict__ v, int hd) {
  __shared__ float red[4];
  const int hbase = blockIdx.x * hd;
  float val = v[hbase + threadIdx.x];
  float s = val * val;
  for (int o = 16; o > 0; o >>= 1) s += __shfl_down(s, o, 32);
  if ((threadIdx.x & 31) == 0) red[threadIdx.x >> 5] = s;
  __syncthreads();
  float tot = red[0] + red[1] + red[2] + red[3];
  v[hbase + threadIdx.x] = val * rsqrtf(tot / (float)hd + EPS_);
}

__global__ void k_finalize(const float* __restrict__ x, const float* __restrict__ g,
                           const float* __restrict__ inv, float* __restrict__ out, int n) {
  int i = blockIdx.x * blockDim.x + threadIdx.x;
  if (i < n) out[i] = x[i] * inv[0] * g[i];
}

// argmax with first-index tie-break; out2[0]=(float)token, out2[1]=fp16-rounded logit
__global__ void k_argmax(const float* __restrict__ logits, int n, float* __restrict__ out2) {
  __shared__ float sv[32];
  __shared__ int si[32];
  float best = -3.4e38f;
  int bi = 0x7fffffff;
  for (int i = threadIdx.x; i < n; i += blockDim.x) {
    float v = logits[i];
    if (v > best) { best = v; bi = i; }  // strided ascending -> keeps first max
  }
  for (int o = 16; o > 0; o >>= 1) {
    float ov = __shfl_down(best, o, 32);
    int oi = __shfl_down(bi, o, 32);
    if (ov > best || (ov == best && oi < bi)) { best = ov; bi = oi; }
  }
  const int wid = threadIdx.x >> 5;
  if ((threadIdx.x & 31) == 0) { sv[wid] = best; si[wid] = bi; }
  __syncthreads();
  if (threadIdx.x < 32) {
    const int nw = (blockDim.x + 31) >> 5;
    float v2 = (threadIdx.x < nw) ? sv[threadIdx.x] : -3.4e38f;
    int i2 = (threadIdx.x < nw) ? si[threadIdx.x] : 0x7fffffff;
    for (int o = 16; o > 0; o >>= 1) {
      float ov = __shfl_down(v2, o, 32);
      int oi = __shfl_down(i2, o, 32);
      if (ov > v2 || (ov == v2 && oi < i2)) { v2 = ov; i2 = oi; }
    }
    if (threadIdx.x == 0) {
      out2[0] = (float)i2;
      out2[1] = __half2float(__float2half_rn(v2));
    }
  }
}

extern "C" void kernel_launch(void* const* d_in, const int* in_sizes, int n_in,
                              void* d_out, int out_size, void* d_ws, size_t ws_size,
                              hipStream_t stream) {
  (void)in_sizes; (void)n_in; (void)out_size; (void)ws_size;
  const float* h_prev = (const float*)d_in[0];
  const float* e_next = (const float*)d_in[1];
  const float* W_in_h = (const float*)d_in[2];
  const float* W_in_e = (const float*)d_in[3];
  const float* g_attn = (const float*)d_in[4];
  // d_in[5]=W_q, d_in[6]=W_k, d_in[8]=g_q, d_in[9]=g_k: dead in output path
  const float* W_v    = (const float*)d_in[7];
  const float* W_o    = (const float*)d_in[10];
  const float* g_ffn  = (const float*)d_in[11];
  const float* W_gate = (const float*)d_in[12];
  const float* W_up   = (const float*)d_in[13];
  const float* W_down = (const float*)d_in[14];
  const float* g_fin  = (const float*)d_in[15];
  const float* W_lm   = (const float*)d_in[16];

  float* ws     = (float*)d_ws;
  float* x      = ws;                         // 4096
  float* v      = x + H_;                     // 1024
  float* t      = v + NKV_ * HD_;             // 14336
  float* logits = t + FFN_;                   // 32000
  float* inv    = logits + VOCAB_;            // 3 scalars

  float* h_out = (float*)d_out;               // [0..4095] h_out, [4096]=token, [4097]=logit

  // 1) x = W_in_h @ h_prev + W_in_e @ e_next
  k_dual_gemv<<<H_ / 128, 256, 2 * H_ * sizeof(float), stream>>>(
      W_in_h, W_in_e, h_prev, e_next, x, H_, H_);
  // 2) pre-attn RMS scale
  k_rms_reduce<<<1, 1024, 0, stream>>>(x, H_, inv + 0);
  // 3) v = W_v @ (x * inv0 * g_attn)
  k_gemv<<<(NKV_ * HD_) / 128, 256, H_ * sizeof(float), stream>>>(
      W_v, x, g_attn, inv + 0, nullptr, 0, v, NKV_ * HD_, H_);
  // 4) per-head RMSNormNoScale of v (trivial T=1 attention)
  k_headnorm<<<NKV_, HD_, 0, stream>>>(v, HD_);
  // 5) x += W_o @ repeat(v)  (GQA gather folded into vector preload)
  k_gemv<<<H_ / 128, 256, H_ * sizeof(float), stream>>>(
      W_o, v, nullptr, nullptr, x, 1, x, H_, H_);
  // 6) pre-ffn RMS scale
  k_rms_reduce<<<1, 1024, 0, stream>>>(x, H_, inv + 1);
  // 7) t = silu(W_gate @ hf) * (W_up @ hf)
  k_gateup<<<FFN_ / 128, 256, H_ * sizeof(float), stream>>>(
      W_gate, W_up, x, g_ffn, inv + 1, t, FFN_, H_);
  // 8) x += W_down @ t
  k_gemv<<<H_ / 128, 256, FFN_ * sizeof(float), stream>>>(
      W_down, t, nullptr, nullptr, x, 0, x, H_, FFN_);
  // 9) final RMS scale
  k_rms_reduce<<<1, 1024, 0, stream>>>(x, H_, inv + 2);
  // 10) h_out = x * inv2 * g_final  -> d_out[0..H-1]
  k_finalize<<<H_ / 256, 256, 0, stream>>>(x, g_fin, inv + 2, h_out, H_);
  // 11) logits = W_lm @ h_out
  k_gemv<<<VOCAB_ / 128, 256, H_ * sizeof(float), stream>>>(
      W_lm, h_out, nullptr, nullptr, nullptr, 0, logits, VOCAB_, H_);
  // 12) token/logit
  k_argmax<<<1, 1024, 0, stream>>>(logits, VOCAB_, h_out + H_);
}